// rODTForestConstruction_10282151707546
// MI455X (gfx1250) — compile-verified
//
#include <hip/hip_runtime.h>
#include <hip/hip_bf16.h>

// Problem constants (from reference)
#define B_       128
#define C_       2048
#define NRODT_   512
#define D_       4
#define NEST_    128
#define NFOREST_ 100
#define NHID_    128
#define EPS_     1e-5f

#define PSTRIDE  132   // P row stride (floats): 132 -> bank walk of 4/lane, conflict-free b64
#define EGSTRIDE 288   // Eg K-pair row stride (floats): 288 % 64 == 32 -> halves hit disjoint banks

typedef __attribute__((ext_vector_type(2))) float v2f;
typedef __attribute__((ext_vector_type(8))) float v8f;

#if defined(__AMDGCN__) && __has_builtin(__builtin_amdgcn_global_load_async_to_lds_b32) && \
    __has_builtin(__builtin_amdgcn_s_wait_asynccnt)
#define HAVE_ASYNC_LDS 1
typedef __attribute__((address_space(1))) int g_int_t;   // global (AS1) int
typedef __attribute__((address_space(3))) int l_int_t;   // LDS (AS3) int
#else
#define HAVE_ASYNC_LDS 0
#endif

// ---------------------------------------------------------------------------
// Kernel 1: per-(batch, group) phi_2 pipeline.
// GN groups == conv groups == 512, each of D_=4 channels, so the whole
// GN -> 4x4 matmul -> ReLU -> GN -> 4x1 matmul chain is local to one thread.
// ---------------------------------------------------------------------------
__global__ __launch_bounds__(256) void phi2_kernel(
    const float* __restrict__ O,
    const float* __restrict__ g1w, const float* __restrict__ g1b,
    const float* __restrict__ W1,  const float* __restrict__ b1,
    const float* __restrict__ g2w, const float* __restrict__ g2b,
    const float* __restrict__ W2,  const float* __restrict__ b2,
    float* __restrict__ wbuf)      // [B_, NRODT_]
{
    const int t = blockIdx.x * 256 + threadIdx.x;   // 0 .. B_*NRODT_-1
    const int b = t >> 9;
    const int g = t & (NRODT_ - 1);

    const float* xp = O + ((size_t)b * C_ + g * 4);
    float x[4];
    x[0] = xp[0]; x[1] = xp[1]; x[2] = xp[2]; x[3] = xp[3];

    float mu = 0.25f * (x[0] + x[1] + x[2] + x[3]);
    float var = 0.25f * (x[0]*x[0] + x[1]*x[1] + x[2]*x[2] + x[3]*x[3]) - mu*mu;
    float rs = rsqrtf(var + EPS_);
    float y[4];
    #pragma unroll
    for (int d = 0; d < 4; ++d)
        y[d] = (x[d] - mu) * rs * g1w[g*4 + d] + g1b[g*4 + d];

    float z[4];
    #pragma unroll
    for (int e = 0; e < 4; ++e) {
        float acc = b1[g*4 + e];
        #pragma unroll
        for (int d = 0; d < 4; ++d)
            acc += y[d] * W1[g*16 + d*4 + e];
        z[e] = fmaxf(acc, 0.0f);
    }

    float mu2 = 0.25f * (z[0] + z[1] + z[2] + z[3]);
    float var2 = 0.25f * (z[0]*z[0] + z[1]*z[1] + z[2]*z[2] + z[3]*z[3]) - mu2*mu2;
    float rs2 = rsqrtf(var2 + EPS_);

    float wv = b2[g];
    #pragma unroll
    for (int d = 0; d < 4; ++d) {
        float zn = (z[d] - mu2) * rs2 * g2w[g*4 + d] + g2b[g*4 + d];
        wv += zn * W2[g*4 + d];
    }
    wbuf[(size_t)b * NRODT_ + g] = wv;
}

// ---------------------------------------------------------------------------
// Kernel 2: one block per forest f.
//  LDS: P  [128 x PSTRIDE]           softmaxed weights, rows = batch, cols = estimator
//       Eg [64 x EGSTRIDE]           gathered E rows, K-pair interleaved:
//                                    element (k, n) at Eg[(k>>1)*EGSTRIDE + 2n + (k&1)]
//                                    -> B fragments are single aligned ds_load_b64
//       idxs[128]                    gather indices (async-staged when available)
//  GEMM: 8 waves x 16-row strips, V_WMMA_F32_16X16X4_F32, K=128, fp32 end-to-end.
// ---------------------------------------------------------------------------
__global__ __launch_bounds__(256) void forest_kernel(
    const float* __restrict__ wbuf,   // [B_, NRODT_]
    const float* __restrict__ E,      // [NRODT_, NHID_]
    const int*   __restrict__ swr,    // [NFOREST_, NEST_]
    float*       __restrict__ out)    // [B_, NFOREST_, NHID_]
{
    extern __shared__ float lds[];
    float* P    = lds;                                   // 128 * PSTRIDE floats
    float* Eg   = lds + 128 * PSTRIDE;                   // 64 * EGSTRIDE floats
    int*   idxs = (int*)(lds + 128 * PSTRIDE + 64 * EGSTRIDE); // 128 ints

    const int f = blockIdx.x;
    const int t = threadIdx.x;

    // ---- stage gather indices (CDNA5 async global->LDS when available) ----
#if HAVE_ASYNC_LDS
    if (t < NEST_) {
        __builtin_amdgcn_global_load_async_to_lds_b32(
            (g_int_t*)(swr + (size_t)f * NEST_ + t),
            (l_int_t*)(idxs + t),
            0, 0);
    }
    __builtin_amdgcn_s_wait_asynccnt(0);
#else
    if (t < NEST_) idxs[t] = swr[(size_t)f * NEST_ + t];
#endif
    __syncthreads();

    // ---- stage gathered E rows, K-pair interleaved ----
    // i = k2*128 + n : threads sweep n fastest -> coalesced 512B global rows
    for (int i = t; i < (NEST_ / 2) * NHID_; i += 256) {
        const int k2 = i >> 7;
        const int n  = i & (NHID_ - 1);
        v2f v;
        v.x = E[(size_t)idxs[2 * k2]     * NHID_ + n];
        v.y = E[(size_t)idxs[2 * k2 + 1] * NHID_ + n];
        *(v2f*)(Eg + k2 * EGSTRIDE + 2 * n) = v;   // 8B-aligned
    }

    // ---- softmax over estimators: 2 threads per batch row, shfl_xor combine ----
    {
        const int row  = t >> 1;       // batch index b
        const int half = t & 1;        // which 64-element half
        const float* wrow = wbuf + (size_t)row * NRODT_;
        __builtin_prefetch(wrow, 0, 1);            // global_prefetch
        float v[64];
        #pragma unroll
        for (int j = 0; j < 64; ++j)
            v[j] = wrow[idxs[half * 64 + j]];
        float m = -3.4e38f;
        #pragma unroll
        for (int j = 0; j < 64; ++j) m = fmaxf(m, v[j]);
        m = fmaxf(m, __shfl_xor(m, 1, 32));        // wave32 partner exchange
        float s = 0.0f;
        #pragma unroll
        for (int j = 0; j < 64; ++j) { v[j] = __expf(v[j] - m); s += v[j]; }
        s += __shfl_xor(s, 1, 32);
        const float inv = 1.0f / s;
        #pragma unroll
        for (int j = 0; j < 64; ++j)
            P[row * PSTRIDE + half * 64 + j] = v[j] * inv;
    }
    __syncthreads();

    // ---- GEMM: F_f[128x128] = P[128x128] @ Eg[128x128] via fp32 WMMA ----
    const int lane = t & 31;
    const int wave = t >> 5;           // 8 waves: wave owns rows m0..m0+15
    const int m0   = wave * 16;
    const int lr   = lane & 15;
    const int hi   = lane >> 4;        // 0: K pair {k,k+1}, 1: K pair {k+2,k+3}

    v8f acc[8] = {};                   // 8 column tiles x 8 VGPRs
    for (int k0 = 0; k0 < NEST_; k0 += 4) {
        const int kb = k0 + 2 * hi;            // even
        const int k2 = (k0 >> 1) + hi;         // K-pair row in Eg
        // A fragment (16x4 f32): one ds_load_b64, K={kb, kb+1}
        v2f a = *(const v2f*)(P + (m0 + lr) * PSTRIDE + kb);
        #pragma unroll
        for (int tn = 0; tn < 8; ++tn) {
            // B fragment (4x16 f32): one ds_load_b64 from the interleaved pair row
            v2f bb = *(const v2f*)(Eg + k2 * EGSTRIDE + 2 * (tn * 16 + lr));
            acc[tn] = __builtin_amdgcn_wmma_f32_16x16x4_f32(
                /*neg_a=*/false, a, /*neg_b=*/false, bb,
                /*c_mod=*/(short)0, acc[tn],
                /*reuse_a=*/false, /*reuse_b=*/false);
        }
    }

    // ---- store: C/D layout -> VGPR r holds row m0 + r + 8*(lane>=16) ----
    const int rowbase = m0 + hi * 8;
    #pragma unroll
    for (int tn = 0; tn < 8; ++tn) {
        const int col = tn * 16 + lr;
        #pragma unroll
        for (int r = 0; r < 8; ++r)
            out[(size_t)(rowbase + r) * (NFOREST_ * NHID_) + (size_t)f * NHID_ + col] = acc[tn][r];
    }
}

// ---------------------------------------------------------------------------
extern "C" void kernel_launch(void* const* d_in, const int* in_sizes, int n_in,
                              void* d_out, int out_size, void* d_ws, size_t ws_size,
                              hipStream_t stream) {
    const float* O   = (const float*)d_in[0];
    const float* g1w = (const float*)d_in[1];
    const float* g1b = (const float*)d_in[2];
    const float* W1  = (const float*)d_in[3];
    const float* b1  = (const float*)d_in[4];
    const float* g2w = (const float*)d_in[5];
    const float* g2b = (const float*)d_in[6];
    const float* W2  = (const float*)d_in[7];
    const float* b2  = (const float*)d_in[8];
    const float* E   = (const float*)d_in[9];
    const int*   swr = (const int*)d_in[10];

    float* wbuf = (float*)d_ws;        // [B_, NRODT_] = 256 KB scratch
    float* out  = (float*)d_out;       // [B_, NFOREST_, NHID_]

    phi2_kernel<<<(B_ * NRODT_) / 256, 256, 0, stream>>>(
        O, g1w, g1b, W1, b1, g2w, g2b, W2, b2, wbuf);

    const size_t ldsBytes =
        (size_t)(128 * PSTRIDE + 64 * EGSTRIDE) * sizeof(float) + NEST_ * sizeof(int); // ~138 KB (<320 KB/WGP, 2 blocks/WGP)
    (void)hipFuncSetAttribute((const void*)forest_kernel,
                              hipFuncAttributeMaxDynamicSharedMemorySize,
                              (int)ldsBytes);
    forest_kernel<<<NFOREST_, 256, ldsBytes, stream>>>(wbuf, E, swr, out);
}